// RELAX_61512521613458
// MI455X (gfx1250) — compile-verified
//
#include <hip/hip_runtime.h>
#include <hip/hip_bf16.h>
#include <math.h>

// ---------------- constants ----------------
#define H_IMG   224
#define W_IMG   224
#define CELLS   7
#define D_FEAT  512
#define NSAMP   3000            // 30 batches * 100
#define K_RAW   147             // 3*7*7
#define K_PAD   160             // K padded to multiple of 4 (and 32)
#define M_PAD   3008            // 3001 rows padded to multiple of 16
#define CHUNK   100             // samples staged into LDS per scan chunk

typedef __attribute__((ext_vector_type(2))) float v2f;
typedef __attribute__((ext_vector_type(8))) float v8f;

// ------------- shared helpers: mask = shifted-crop of reflect-padded bilinear upsample -------------
// t = i + shift - 3  (range [-3, 226]); reflect to [0,223]; bilinear coord into 7-cell grid.
__device__ __forceinline__ void axis_params(int t, int& i0, float& fr) {
    int p = (t < 0) ? -t : ((t > 223) ? (446 - t) : t);
    float f = ((float)p + 0.5f) * (1.0f / 32.0f) - 0.5f;       // jax.image.resize linear coord
    f = fminf(fmaxf(f, 0.0f), 6.0f);                            // edge renormalization == clamp
    int i = (int)f;                                             // f >= 0: trunc == floor
    if (i > 5) i = 5;
    i0 = i;
    fr = f - (float)i;
}

__device__ __forceinline__ float mask_val(const float* __restrict__ g49,
                                          int sx, int sy, int i, int j) {
    int iu, iv; float fu, fv;
    axis_params(i + sx - 3, iu, fu);
    axis_params(j + sy - 3, iv, fv);
    const float* r0 = g49 + iu * 7 + iv;
    float m0 = r0[0] + fv * (r0[1] - r0[0]);
    float m1 = r0[7] + fv * (r0[8] - r0[7]);
    return m0 + fu * (m1 - m0);
}

// ---------------- init kernels ----------------
__global__ void zero_kernel(float* __restrict__ p, int n) {
    int i = blockIdx.x * 256 + threadIdx.x;
    if (i < n) p[i] = 0.0f;
}

__global__ void pad_wenc_kernel(const float* __restrict__ W, float* __restrict__ Wp) {
    int idx = blockIdx.x * 256 + threadIdx.x;
    if (idx >= K_PAD * D_FEAT) return;
    int k = idx >> 9;           // /512
    int n = idx & 511;
    Wp[idx] = (k < K_RAW) ? W[k * D_FEAT + n] : 0.0f;
}

// ---------------- kernel A: masked block-pool features ----------------
// one block per sample; sample NSAMP (3000) is the unmasked h_star row.
// feats[s][c*49 + cell] = (1/1024) * sum_{32x32 block} x[c] * mask
__global__ void feats_kernel(const float* __restrict__ x,
                             const float* __restrict__ grids,
                             const int* __restrict__ sxA,
                             const int* __restrict__ syA,
                             float* __restrict__ feats) {
    int s   = blockIdx.x;                    // 0..3000
    int tid = threadIdx.x;
    __shared__ float gsh[49];
    __shared__ int   shXY[2];
    __shared__ float red[196 * 3];
    bool star = (s == NSAMP);
    if (!star) {
        if (tid < 49) gsh[tid] = grids[s * 49 + tid];
        if (tid == 49) shXY[0] = sxA[s];
        if (tid == 50) shXY[1] = syA[s];
    }
    __syncthreads();
    if (tid < 196) {
        int cell = tid >> 2, part = tid & 3;
        int cr = cell / 7, cc = cell % 7;
        float a0 = 0.f, a1 = 0.f, a2 = 0.f;
        int sx = star ? 0 : shXY[0];
        int sy = star ? 0 : shXY[1];
        for (int q = 0; q < 256; ++q) {
            int idx = part * 256 + q;
            int i = cr * 32 + (idx >> 5);
            int j = cc * 32 + (idx & 31);
            float m = star ? 1.0f : mask_val(gsh, sx, sy, i, j);
            int off = i * W_IMG + j;
            a0 += x[off] * m;
            a1 += x[50176 + off] * m;
            a2 += x[100352 + off] * m;
        }
        red[tid * 3 + 0] = a0;
        red[tid * 3 + 1] = a1;
        red[tid * 3 + 2] = a2;
    }
    __syncthreads();
    if (tid < 196 && (tid & 3) == 0) {
        int cell = tid >> 2;
        #pragma unroll
        for (int c = 0; c < 3; ++c) {
            float v = red[tid * 3 + c] + red[(tid + 1) * 3 + c] +
                      red[(tid + 2) * 3 + c] + red[(tid + 3) * 3 + c];
            feats[s * K_PAD + c * 49 + cell] = v * (1.0f / 1024.0f);
        }
    }
}

// ---------------- kernel B: fp32 WMMA GEMM  C[3008x512] = A[3008x160] @ B[160x512] ----------------
// one 16x16 tile per wave, V_WMMA_F32_16X16X4_F32, K loop of 40.
// A layout (16x4 f32): lanes 0-15 -> K pair {0,1}; lanes 16-31 -> K pair {2,3}; M = lane&15
// B layout (4x16 f32): symmetric; N = lane&15
// C/D (16x16 f32, 8 VGPRs): VGPR r -> M = r + (lane<16 ? 0 : 8), N = lane&15
__global__ void gemm_wmma_kernel(const float* __restrict__ A,
                                 const float* __restrict__ B,
                                 float* __restrict__ C) {
    const int tiles_n = D_FEAT / 16;                       // 32
    int wave = (blockIdx.x * blockDim.x + threadIdx.x) >> 5;
    int lane = threadIdx.x & 31;
    int tm = wave / tiles_n;                               // 0..187
    int tn = wave % tiles_n;                               // 0..31
    int mrow  = tm * 16 + (lane & 15);
    int ncol  = tn * 16 + (lane & 15);
    int khalf = (lane >> 4) << 1;                          // 0 or 2
    v8f acc = {};
    const float* arow = A + mrow * K_PAD;
    #pragma unroll 4
    for (int k0 = 0; k0 < K_PAD; k0 += 4) {
        v2f a = *(const v2f*)(arow + k0 + khalf);          // 8B aligned: K_PAD even, k0+khalf even
        v2f b;
        b.x = B[(k0 + khalf) * D_FEAT + ncol];
        b.y = B[(k0 + khalf + 1) * D_FEAT + ncol];
        acc = __builtin_amdgcn_wmma_f32_16x16x4_f32(
                  /*neg_a=*/false, a, /*neg_b=*/false, b,
                  /*c_mod=*/(short)0, acc,
                  /*reuse_a=*/false, /*reuse_b=*/false);
    }
    int mbase = tm * 16 + ((lane >> 4) << 3);
    #pragma unroll
    for (int r = 0; r < 8; ++r)
        C[(mbase + r) * D_FEAT + ncol] = acc[r];
}

// ---------------- kernel C: cosine sims vs h_star ----------------
__global__ void sims_kernel(const float* __restrict__ Hm, float* __restrict__ sims) {
    int s   = blockIdx.x;                     // 0..2999
    int tid = threadIdx.x;
    const float* hs = Hm + s * D_FEAT;
    const float* ht = Hm + NSAMP * D_FEAT;    // h_star row
    float d = 0.f, n1 = 0.f, n2 = 0.f;
    for (int k = tid; k < D_FEAT; k += 256) {
        float a = ht[k], b = hs[k];
        d += a * b; n1 += a * a; n2 += b * b;
    }
    __shared__ float sd[256], s1[256], s2[256];
    sd[tid] = d; s1[tid] = n1; s2[tid] = n2;
    __syncthreads();
    for (int st = 128; st > 0; st >>= 1) {
        if (tid < st) { sd[tid] += sd[tid + st]; s1[tid] += s1[tid + st]; s2[tid] += s2[tid + st]; }
        __syncthreads();
    }
    if (tid == 0)
        sims[s] = sd[0] / fmaxf(sqrtf(s1[0]) * sqrtf(s2[0]), 1e-8f);
}

// ---------------- kernel D: per-pixel ordered Welford scan over 3000 samples ----------------
__global__ void scan_kernel(const float* __restrict__ grids,
                            const int* __restrict__ sxA,
                            const int* __restrict__ syA,
                            const float* __restrict__ sims,
                            float* __restrict__ out) {
    __shared__ float sh_g[CHUNK * 49];
    __shared__ float sh_s[CHUNK];
    __shared__ int   sh_x[CHUNK];
    __shared__ int   sh_y[CHUNK];
    int tid = threadIdx.x;
    int p = blockIdx.x * 256 + tid;           // 196*256 == 50176 == 224*224 exactly
    int i = p / W_IMG, j = p % W_IMG;
    float R = 0.0f, U = 0.0f, sow = 1e-10f;
    for (int c0 = 0; c0 < NSAMP; c0 += CHUNK) {
        __syncthreads();
        for (int t = tid; t < CHUNK * 49; t += 256) sh_g[t] = grids[c0 * 49 + t];
        for (int t = tid; t < CHUNK; t += 256) {
            sh_s[t] = sims[c0 + t];
            sh_x[t] = sxA[c0 + t];
            sh_y[t] = syA[c0 + t];
        }
        __syncthreads();
        for (int t = 0; t < CHUNK; ++t) {
            float m = mask_val(&sh_g[t * 49], sh_x[t], sh_y[t], i, j);
            float s = sh_s[t];
            sow += m;
            float Rn = R + m * (s - R) / sow;
            U += (s - Rn) * (s - R) * m;
            R = Rn;
        }
    }
    out[p] = R;
    out[H_IMG * W_IMG + p] = U / (sow - 1.0f);
}

// ---------------- launch ----------------
extern "C" void kernel_launch(void* const* d_in, const int* in_sizes, int n_in,
                              void* d_out, int out_size, void* d_ws, size_t ws_size,
                              hipStream_t stream) {
    const float* x     = (const float*)d_in[0];   // [1,3,224,224]
    const float* Wenc  = (const float*)d_in[1];   // [147,512]
    const float* grids = (const float*)d_in[2];   // [30,100,1,7,7] -> flat [3000,49]
    const int*   sx    = (const int*)d_in[3];     // [3000]
    const int*   sy    = (const int*)d_in[4];     // [3000]
    float* out = (float*)d_out;                   // [2,224,224]

    float* ws    = (float*)d_ws;
    float* feats = ws;                             // M_PAD*K_PAD      = 481,280 f
    float* Wp    = feats + (size_t)M_PAD * K_PAD;  // K_PAD*D_FEAT     =  81,920 f
    float* Hm    = Wp    + (size_t)K_PAD * D_FEAT; // M_PAD*D_FEAT     = 1,540,096 f
    float* sims  = Hm    + (size_t)M_PAD * D_FEAT; // 3000 f           (~8.4 MB total)

    int nfeat = M_PAD * K_PAD;
    zero_kernel<<<(nfeat + 255) / 256, 256, 0, stream>>>(feats, nfeat);
    pad_wenc_kernel<<<(K_PAD * D_FEAT + 255) / 256, 256, 0, stream>>>(Wenc, Wp);
    feats_kernel<<<NSAMP + 1, 256, 0, stream>>>(x, grids, sx, sy, feats);
    // 188 M-tiles * 32 N-tiles = 6016 waves; 8 waves/block -> 752 blocks (exact, no guard)
    gemm_wmma_kernel<<<752, 256, 0, stream>>>(feats, Wp, Hm);
    sims_kernel<<<NSAMP, 256, 0, stream>>>(Hm, sims);
    scan_kernel<<<196, 256, 0, stream>>>(grids, sx, sy, sims, out);
}